// MHSA2D_5901285065305
// MI455X (gfx1250) — compile-verified
//
#include <hip/hip_runtime.h>
#include <math.h>

typedef __attribute__((ext_vector_type(2))) float v2f;
typedef __attribute__((ext_vector_type(8))) float v8f;

#define DIMC 768
#define NTOK 1024
#define NHD  8
#define HD   96
#define NB   16
#define SROW 1028   // LDS row stride (floats): 1028%8==4 -> bank-conflict-free phases

static __device__ __forceinline__ v8f wmma4(v2f a, v2f b, v8f c) {
    return __builtin_amdgcn_wmma_f32_16x16x4_f32(false, a, false, b, (short)0, c,
                                                 false, false);
}

// -------------------------------------------------------------------------
// Y[b] = W (O x 768) * X[b] (768 x 1024) + bias, fp32 WMMA.
// 8 waves / block, each wave owns a 32x32 output tile (2x2 of 16x16 frags).
// A frag (16x4): lane l: row M = l&15, holds K = 2*(l>>4)+{0,1}
// B frag (4x16): lane l: col N = l&15, holds K = 2*(l>>4)+{0,1}
// C/D frag     : VGPR j, lane l: M = j + 8*(l>>4), N = l&15
// -------------------------------------------------------------------------
__global__ __launch_bounds__(256) void gemm_bias_f32(
    const float* __restrict__ W, const float* __restrict__ bias,
    const float* __restrict__ X, float* __restrict__ Y, int Orows)
{
    constexpr int K = DIMC;
    constexpr int N = NTOK;

    const int wave = threadIdx.x >> 5;
    const int lane = threadIdx.x & 31;
    const int half = lane >> 4;
    const int r    = lane & 15;

    const int tilesN = N >> 5;                 // 32-wide tiles
    const int tile = blockIdx.x * 8 + wave;
    const int ot = tile / tilesN;
    const int nt = tile - ot * tilesN;
    const int o0 = ot << 5;
    const int n0 = nt << 5;

    const float* __restrict__ Xb = X + (size_t)blockIdx.y * (size_t)K * N;
    float*       __restrict__ Yb = Y + (size_t)blockIdx.y * (size_t)Orows * N;

    const float* w0 = W + (size_t)(o0 + r) * K + 2 * half;       // A rows o0..o0+15
    const float* w1 = w0 + (size_t)16 * K;                        // A rows o0+16..o0+31
    const float* x0 = Xb + (size_t)(2 * half) * N + n0 + r;       // B cols n0..n0+15
                                                                  // B cols +16: x0+16

    v8f acc00 = {}, acc01 = {}, acc10 = {}, acc11 = {};

    for (int kc = 0; kc < K; kc += 16) {
        if (kc + 32 < K)                       // stream-prefetch X (global_prefetch_b8)
            __builtin_prefetch(x0 + (size_t)(kc + 32) * N, 0, 1);
#pragma unroll
        for (int kk = 0; kk < 16; kk += 4) {
            const int k = kc + kk;
            v2f a0 = {w0[k], w0[k + 1]};
            v2f a1 = {w1[k], w1[k + 1]};
            v2f b0 = {x0[(size_t)k * N],      x0[(size_t)(k + 1) * N]};
            v2f b1 = {x0[(size_t)k * N + 16], x0[(size_t)(k + 1) * N + 16]};
            acc00 = wmma4(a0, b0, acc00);
            acc01 = wmma4(a0, b1, acc01);
            acc10 = wmma4(a1, b0, acc10);
            acc11 = wmma4(a1, b1, acc11);
        }
    }

#pragma unroll
    for (int j = 0; j < 8; ++j) {
        const int oA = o0 + j + 8 * half;
        const int oB = oA + 16;
        const float bA = bias[oA], bB = bias[oB];
        Yb[(size_t)oA * N + n0 + r]      = acc00[j] + bA;
        Yb[(size_t)oA * N + n0 + 16 + r] = acc01[j] + bA;
        Yb[(size_t)oB * N + n0 + r]      = acc10[j] + bB;
        Yb[(size_t)oB * N + n0 + 16 + r] = acc11[j] + bB;
    }
}

// -------------------------------------------------------------------------
// Attention for one (b, h, 16-row block of n).
// qkv layout: [b][o][n], o = part*768 + h*96 + d.
// Phase 1: S[16][1024] = (q^T k) * hd^-0.5 -> LDS; q A-frags hoisted (48 VGPRs)
// Phase 2: row softmax (max-sub, clip -88, normalize), wave32 shuffles
// Phase 3: O^T = P * v^T via WMMA; each LDS A-frag feeds two d-tiles
// -------------------------------------------------------------------------
__global__ __launch_bounds__(32) void attn_f32(
    const float* __restrict__ qkv, float* __restrict__ O)
{
    extern __shared__ float S[];   // 16 * SROW floats

    const int lane = threadIdx.x;
    const int half = lane >> 4;
    const int r    = lane & 15;

    const int n0 = blockIdx.x << 4;
    const int b  = blockIdx.y >> 3;
    const int h  = blockIdx.y & 7;

    const float* base = qkv + (size_t)b * (3 * DIMC) * NTOK + (size_t)h * HD * NTOK;
    const float* qb = base;
    const float* kb = base + (size_t)DIMC * NTOK;
    const float* vb = base + (size_t)(2 * DIMC) * NTOK;

    const float s2 = 0.10206207261596575f;  // 96^-0.5 == (hd^-0.25)^2

    // ---- hoist q A-fragments for this 16-row block: 24 k-steps of 4 ----
    v2f qa[HD / 4];
#pragma unroll
    for (int t = 0; t < HD / 4; ++t) {
        const float* qp = qb + (size_t)(4 * t + 2 * half) * NTOK + n0 + r;
        qa[t] = {qp[0], qp[NTOK]};
    }

    // ---- Phase 1: scores, two 16-wide m-tiles per pass (shared A) ----
    for (int m0 = 0; m0 < NTOK; m0 += 32) {
        v8f acc0 = {}, acc1 = {};
#pragma unroll
        for (int t = 0; t < HD / 4; ++t) {
            const float* kp = kb + (size_t)(4 * t + 2 * half) * NTOK + m0 + r;
            v2f b0 = {kp[0],  kp[NTOK]};
            v2f b1 = {kp[16], kp[NTOK + 16]};
            acc0 = wmma4(qa[t], b0, acc0);
            acc1 = wmma4(qa[t], b1, acc1);
        }
#pragma unroll
        for (int j = 0; j < 8; ++j) {
            const int row = (j + 8 * half) * SROW;
            S[row + m0 + r]      = acc0[j] * s2;
            S[row + m0 + 16 + r] = acc1[j] * s2;
        }
    }
    __syncthreads();

    // ---- Phase 2: softmax over each of the 16 rows ----
    for (int row = 0; row < 16; ++row) {
        float* Sr = &S[row * SROW];
        float mx = -3.402823466e38f;
        for (int c = lane; c < NTOK; c += 32) mx = fmaxf(mx, Sr[c]);
#pragma unroll
        for (int off = 16; off >= 1; off >>= 1) mx = fmaxf(mx, __shfl_xor(mx, off, 32));

        float sum = 0.f;
        for (int c = lane; c < NTOK; c += 32) {
            float p = __expf(fmaxf(Sr[c] - mx, -88.0f));
            Sr[c] = p;
            sum += p;
        }
#pragma unroll
        for (int off = 16; off >= 1; off >>= 1) sum += __shfl_xor(sum, off, 32);

        const float inv = 1.0f / sum;
        for (int c = lane; c < NTOK; c += 32) Sr[c] *= inv;
    }
    __syncthreads();

    // ---- Phase 3: out = P * v^T, two 16-wide d-tiles per pass (shared A) ----
    float* Ob = O + ((size_t)b * DIMC + (size_t)h * HD) * NTOK;
    for (int d0 = 0; d0 < HD; d0 += 32) {
        v8f acc0 = {}, acc1 = {};
        for (int m0 = 0; m0 < NTOK; m0 += 4) {
            const float* sp = &S[r * SROW + m0 + 2 * half];
            v2f a = {sp[0], sp[1]};
            const float* vp  = vb + (size_t)(d0 + r) * NTOK + m0 + 2 * half;
            const float* vp1 = vp + (size_t)16 * NTOK;
            v2f b0 = {vp[0],  vp[1]};
            v2f b1 = {vp1[0], vp1[1]};
            acc0 = wmma4(a, b0, acc0);
            acc1 = wmma4(a, b1, acc1);
        }
#pragma unroll
        for (int j = 0; j < 8; ++j) {
            const int tok = n0 + j + 8 * half;
            Ob[(size_t)(d0 + r) * NTOK + tok]      = acc0[j];
            Ob[(size_t)(d0 + 16 + r) * NTOK + tok] = acc1[j];
        }
    }
}

// -------------------------------------------------------------------------
extern "C" void kernel_launch(void* const* d_in, const int* in_sizes, int n_in,
                              void* d_out, int out_size, void* d_ws, size_t ws_size,
                              hipStream_t stream) {
    const float* x      = (const float*)d_in[0];  // (16,768,32,32) == (16,768,1024)
    const float* w_qkv  = (const float*)d_in[1];  // (2304,768)
    const float* b_qkv  = (const float*)d_in[2];  // (2304,)
    const float* w_proj = (const float*)d_in[3];  // (768,768)
    const float* b_proj = (const float*)d_in[4];  // (768,)
    float* out = (float*)d_out;                   // (16,768,1024)

    // workspace carve: qkv (151 MB) + attention output (50 MB)
    float* qkv  = (float*)d_ws;
    float* obuf = qkv + (size_t)NB * 3 * DIMC * NTOK;

    // 1) qkv = w_qkv @ x + b_qkv   (2304/32 * 1024/32 = 2304 tiles, 8 waves/blk)
    {
        dim3 grid((3 * DIMC / 32) * (NTOK / 32) / 8, NB);
        gemm_bias_f32<<<grid, 256, 0, stream>>>(w_qkv, b_qkv, x, qkv, 3 * DIMC);
    }
    // 2) attention per (b, h, 16-row block)
    {
        dim3 grid(NTOK / 16, NB * NHD);
        attn_f32<<<grid, 32, 16 * SROW * sizeof(float), stream>>>(qkv, obuf);
    }
    // 3) out = w_proj @ obuf + b_proj
    {
        dim3 grid((DIMC / 32) * (NTOK / 32) / 8, NB);
        gemm_bias_f32<<<grid, 256, 0, stream>>>(w_proj, b_proj, obuf, out, DIMC);
    }
}